// ComplexSSM_42090679501212
// MI455X (gfx1250) — compile-verified
//
#include <hip/hip_runtime.h>
#include <hip/hip_bf16.h>

typedef __attribute__((ext_vector_type(2))) float v2f;
typedef __attribute__((ext_vector_type(8))) float v8f;

namespace {
constexpr int Bb   = 8;
constexpr int Tt   = 4096;
constexpr int Dd   = 1024;
constexpr int Rr   = 512;
constexpr int M_   = Bb * Tt;     // 32768 rows
constexpr int NTOT = 3 * Rr;      // 1536 fused output cols (512 x_r | 1024 yres)
constexpr int CHUNK = 128;
constexpr int NCH   = Tt / CHUNK; // 32
constexpr float EPS = 1e-5f;

constexpr int KB    = 32;         // K-slab per LDS stage
constexpr int NSTG  = Dd / KB;    // 32 stages
constexpr int LDA   = 36;         // padded LDS stride (16B-aligned rows, bank-spread)
}

// global -> LDS async copy, 16 bytes per lane (ASYNCcnt-tracked)
__device__ __forceinline__ void async_copy16(unsigned lds_byte_off,
                                             unsigned long long gaddr) {
    asm volatile("global_load_async_to_lds_b128 %0, %1, off"
                 :: "v"(lds_byte_off), "v"(gaddr) : "memory");
}
__device__ __forceinline__ void wait_async_le6() {
    asm volatile("s_wait_asynccnt 0x6" ::: "memory");
}
__device__ __forceinline__ void wait_async_0() {
    asm volatile("s_wait_asynccnt 0x0" ::: "memory");
}
__device__ __forceinline__ unsigned lds_off(const void* p) {
    return (unsigned)(uintptr_t)p;   // flat LDS addr[31:0] == LDS byte offset
}

// ---------------------------------------------------------------------------
// Kernel 1: fused GEMM  [u @ [W_proj;W_res]^T]  via V_WMMA_F32_16X16X4_F32.
// Block = 256 thr (8 waves). Block tile = 128 rows x 64 cols, wave = 16x64.
// Double-buffered LDS staging fed by global_load_async_to_lds_b128.
// ---------------------------------------------------------------------------
__global__ __launch_bounds__(256) void ssm_gemm_wmma(
    const float* __restrict__ u, const float* __restrict__ Wproj,
    const float* __restrict__ Wres, float* __restrict__ xr,
    float* __restrict__ yres)
{
    __shared__ float Abuf[2][128 * LDA];   // 2 x 18 KB
    __shared__ float Bbuf[2][64 * LDA];    // 2 x  9 KB

    const int tid  = threadIdx.x;
    const int lane = tid & 31;
    const int wave = tid >> 5;
    const int rowBlk = blockIdx.x * 128;
    const int col0   = blockIdx.y * 64;

    const int nl = lane & 15;            // M index (A frag) / N index (B frag)
    const int kb = (lane >> 4) << 1;     // K sub-offset {0,2}

    const float* wmat; int wc0;
    if (col0 < Rr) { wmat = Wproj; wc0 = col0; }
    else           { wmat = Wres;  wc0 = col0 - Rr; }

    // per-stage async copy: A = 1024 16B-chunks (4/thread), B = 512 (2/thread)
    auto issue_stage = [&](int s, int buf) {
        const int kk = s * KB;
        unsigned ab = lds_off(&Abuf[buf][0]);
        unsigned bbf = lds_off(&Bbuf[buf][0]);
        #pragma unroll
        for (int i = 0; i < 4; ++i) {
            int q   = tid + 256 * i;
            int row = q >> 3;
            int kc  = (q & 7) << 2;
            const float* g = u + (size_t)(rowBlk + row) * Dd + (kk + kc);
            async_copy16(ab + (unsigned)(row * LDA + kc) * 4u,
                         (unsigned long long)(uintptr_t)g);
        }
        #pragma unroll
        for (int i = 0; i < 2; ++i) {
            int q   = tid + 256 * i;
            int col = q >> 3;
            int kc  = (q & 7) << 2;
            const float* g = wmat + (size_t)(wc0 + col) * Dd + (kk + kc);
            async_copy16(bbf + (unsigned)(col * LDA + kc) * 4u,
                         (unsigned long long)(uintptr_t)g);
        }
    };

    v8f acc0 = {}, acc1 = {}, acc2 = {}, acc3 = {};

    auto compute_stage = [&](int buf) {
        const float* As  = &Abuf[buf][(wave * 16 + nl) * LDA + kb];
        const float* Bs0 = &Bbuf[buf][( 0 + nl) * LDA + kb];
        const float* Bs1 = &Bbuf[buf][(16 + nl) * LDA + kb];
        const float* Bs2 = &Bbuf[buf][(32 + nl) * LDA + kb];
        const float* Bs3 = &Bbuf[buf][(48 + nl) * LDA + kb];
        #pragma unroll
        for (int k4 = 0; k4 < KB; k4 += 4) {
            v2f a  = *(const v2f*)(As  + k4);
            v2f b0 = *(const v2f*)(Bs0 + k4);
            v2f b1 = *(const v2f*)(Bs1 + k4);
            v2f b2 = *(const v2f*)(Bs2 + k4);
            v2f b3 = *(const v2f*)(Bs3 + k4);
            acc0 = __builtin_amdgcn_wmma_f32_16x16x4_f32(false, a, false, b0, (short)0, acc0, false, false);
            acc1 = __builtin_amdgcn_wmma_f32_16x16x4_f32(false, a, false, b1, (short)0, acc1, false, false);
            acc2 = __builtin_amdgcn_wmma_f32_16x16x4_f32(false, a, false, b2, (short)0, acc2, false, false);
            acc3 = __builtin_amdgcn_wmma_f32_16x16x4_f32(false, a, false, b3, (short)0, acc3, false, false);
        }
    };

    issue_stage(0, 0);
    for (int s = 0; s < NSTG - 1; ++s) {
        issue_stage(s + 1, (s + 1) & 1);   // prefetch next slab into other buffer
        wait_async_le6();                   // stage s complete (next stage in flight)
        __syncthreads();
        compute_stage(s & 1);
        __syncthreads();                    // done with buf before it's re-filled
    }
    wait_async_0();
    __syncthreads();
    compute_stage((NSTG - 1) & 1);

    // C/D layout: VGPR j -> rows (lane<16 ? j : 8+j), col = lane&15
    float* obase; int ldo, oc;
    if (col0 < Rr) { obase = xr;   ldo = Rr;     oc = col0; }
    else           { obase = yres; ldo = 2 * Rr; oc = col0 - Rr; }
    const int mr = rowBlk + wave * 16 + ((lane >> 4) << 3);
    float* d0 = obase + (size_t)mr * ldo + (oc +  0 + nl);
    float* d1 = obase + (size_t)mr * ldo + (oc + 16 + nl);
    float* d2 = obase + (size_t)mr * ldo + (oc + 32 + nl);
    float* d3 = obase + (size_t)mr * ldo + (oc + 48 + nl);
    #pragma unroll
    for (int j = 0; j < 8; ++j) {
        d0[(size_t)j * ldo] = acc0[j];
        d1[(size_t)j * ldo] = acc1[j];
        d2[(size_t)j * ldo] = acc2[j];
        d3[(size_t)j * ldo] = acc3[j];
    }
}

__device__ __forceinline__ void decay_coeffs(const float* lraw, const float* omega,
                                             int r, float& ar, float& ai) {
    float lam = -5.f / (1.f + expf(-lraw[r]));
    float mag = expf(lam);
    float w   = omega[r];
    ar = mag * cosf(w);
    ai = mag * sinf(w);
}

// ---------------------------------------------------------------------------
// Kernel 2 (scan phase A): local complex scan per chunk, h accumulated into
// d_out (which already holds u@W_res^T), chunk-final state to ws.
// ---------------------------------------------------------------------------
__global__ __launch_bounds__(256) void ssm_scan_local(
    const float* __restrict__ x, const float* __restrict__ lraw,
    const float* __restrict__ omega, const float* __restrict__ h0r,
    const float* __restrict__ h0i, float* __restrict__ y,
    float* __restrict__ hend)
{
    const int flat = blockIdx.x * blockDim.x + threadIdx.x;
    const int r = flat % Rr;
    const int c = (flat / Rr) % NCH;
    const int b = flat / (Rr * NCH);

    float ar, ai; decay_coeffs(lraw, omega, r, ar, ai);

    float inj_r = 0.f, inj_i = 0.f;
    if (c == 0) {  // fold initial state into x[0]:  x0 += a * h0
        float h0rv = h0r[b * Rr + r], h0iv = h0i[b * Rr + r];
        inj_r = ar * h0rv - ai * h0iv;
        inj_i = ar * h0iv + ai * h0rv;
    }

    const int t0 = c * CHUNK;
    const float* xp = x + ((size_t)b * Tt + t0) * Rr + r;
    float*       yp = y + ((size_t)b * Tt + t0) * (2 * Rr) + r;

    float hr = 0.f, hi = 0.f;
    for (int k = 0; k < CHUNK; ++k) {
        float xin_r = xp[(size_t)k * Rr];
        float xin_i = 0.f;
        if (k == 0) { xin_r += inj_r; xin_i += inj_i; }  // inj==0 for c>0
        float nhr = ar * hr - ai * hi + xin_r;
        float nhi = ar * hi + ai * hr + xin_i;
        hr = nhr; hi = nhi;
        yp[(size_t)k * (2 * Rr)]      += hr;
        yp[(size_t)k * (2 * Rr) + Rr] += hi;
    }
    size_t ho = (((size_t)b * Rr + r) * NCH + c) * 2;
    hend[ho] = hr; hend[ho + 1] = hi;
}

// ---------------------------------------------------------------------------
// Kernel 3 (scan phase B): carry_{c+1} = a^CHUNK * carry_c + hend_c, carry_0=0.
// Emits final hidden state (carry_{NCH}) into the tail of d_out.
// ---------------------------------------------------------------------------
__global__ __launch_bounds__(256) void ssm_scan_carry(
    const float* __restrict__ lraw, const float* __restrict__ omega,
    const float* __restrict__ hend, float* __restrict__ carry,
    float* __restrict__ finals)
{
    const int flat = blockIdx.x * blockDim.x + threadIdx.x;
    const int r = flat % Rr;
    const int b = flat / Rr;

    float lam = -5.f / (1.f + expf(-lraw[r]));
    float w   = omega[r];
    float Lf  = (float)CHUNK;
    float mgL = expf(lam * Lf);          // |a|^CHUNK
    float aLr = mgL * cosf(w * Lf);
    float aLi = mgL * sinf(w * Lf);

    size_t base = ((size_t)b * Rr + r) * NCH * 2;
    float cr = 0.f, ci = 0.f;
    for (int c = 0; c < NCH; ++c) {
        carry[base + 2 * c]     = cr;
        carry[base + 2 * c + 1] = ci;
        float er = hend[base + 2 * c], ei = hend[base + 2 * c + 1];
        float ncr = aLr * cr - aLi * ci + er;
        float nci = aLr * ci + aLi * cr + ei;
        cr = ncr; ci = nci;
    }
    finals[(size_t)b * Rr + r]                   = cr;  // final_r
    finals[(size_t)Bb * Rr + (size_t)b * Rr + r] = ci;  // final_i
}

// ---------------------------------------------------------------------------
// Kernel 4 (scan phase C): h[t0+k] += a^{k+1} * carry_c.
// ---------------------------------------------------------------------------
__global__ __launch_bounds__(256) void ssm_scan_apply(
    const float* __restrict__ lraw, const float* __restrict__ omega,
    const float* __restrict__ carry, float* __restrict__ y)
{
    const int flat = blockIdx.x * blockDim.x + threadIdx.x;
    const int r = flat % Rr;
    const int c = (flat / Rr) % NCH;
    const int b = flat / (Rr * NCH);
    if (c == 0) return;  // carry_0 == 0; c uniform per wave -> no divergence

    size_t co = (((size_t)b * Rr + r) * NCH + c) * 2;
    float cr = carry[co], ci = carry[co + 1];

    float ar, ai; decay_coeffs(lraw, omega, r, ar, ai);
    float pr = ar, pi = ai;  // a^{k+1}

    float* yp = y + ((size_t)b * Tt + c * CHUNK) * (2 * Rr) + r;
    for (int k = 0; k < CHUNK; ++k) {
        yp[(size_t)k * (2 * Rr)]      += pr * cr - pi * ci;
        yp[(size_t)k * (2 * Rr) + Rr] += pr * ci + pi * cr;
        float npr = pr * ar - pi * ai;
        float npi = pr * ai + pi * ar;
        pr = npr; pi = npi;
    }
}

// ---------------------------------------------------------------------------
// Kernel 5: LayerNorm over the 2R=1024 channel dim, wave32-per-row, in place.
// ---------------------------------------------------------------------------
__global__ __launch_bounds__(256) void ssm_layernorm(
    float* __restrict__ y, const float* __restrict__ gamma,
    const float* __restrict__ beta)
{
    const int lane = threadIdx.x & 31;
    const int wave = threadIdx.x >> 5;
    const size_t row = (size_t)blockIdx.x * 8 + wave;
    float* p = y + row * (2 * Rr);

    float4 v[8];
    float s = 0.f, s2 = 0.f;
    #pragma unroll
    for (int j = 0; j < 8; ++j) {
        v[j] = ((const float4*)p)[j * 32 + lane];
        s  += v[j].x + v[j].y + v[j].z + v[j].w;
        s2 += v[j].x * v[j].x + v[j].y * v[j].y + v[j].z * v[j].z + v[j].w * v[j].w;
    }
    #pragma unroll
    for (int m = 16; m >= 1; m >>= 1) {
        s  += __shfl_xor(s,  m, 32);
        s2 += __shfl_xor(s2, m, 32);
    }
    const float mean = s * (1.f / (2 * Rr));
    const float var  = s2 * (1.f / (2 * Rr)) - mean * mean;
    const float rstd = rsqrtf(var + EPS);

    #pragma unroll
    for (int j = 0; j < 8; ++j) {
        float4 g  = ((const float4*)gamma)[j * 32 + lane];
        float4 bt = ((const float4*)beta)[j * 32 + lane];
        float4 o;
        o.x = g.x * (v[j].x - mean) * rstd + bt.x;
        o.y = g.y * (v[j].y - mean) * rstd + bt.y;
        o.z = g.z * (v[j].z - mean) * rstd + bt.z;
        o.w = g.w * (v[j].w - mean) * rstd + bt.w;
        ((float4*)p)[j * 32 + lane] = o;
    }
}

extern "C" void kernel_launch(void* const* d_in, const int* in_sizes, int n_in,
                              void* d_out, int out_size, void* d_ws, size_t ws_size,
                              hipStream_t stream) {
    const float* u     = (const float*)d_in[0];
    const float* h0r   = (const float*)d_in[1];
    const float* h0i   = (const float*)d_in[2];
    const float* lraw  = (const float*)d_in[3];
    const float* omega = (const float*)d_in[4];
    const float* Wproj = (const float*)d_in[5];
    const float* Wres  = (const float*)d_in[6];
    const float* gamma = (const float*)d_in[7];
    const float* beta  = (const float*)d_in[8];

    float* out    = (float*)d_out;
    float* yres   = out;                              // [B*T, 2R] staging/in-place
    float* finals = out + (size_t)M_ * 2 * Rr;        // final_r | final_i tail

    float* xr    = (float*)d_ws;                      // [B*T, R]  64 MB
    float* hend  = xr + (size_t)M_ * Rr;              // [B,R,NCH,2]  1 MB
    float* carry = hend + (size_t)Bb * Rr * NCH * 2;  // [B,R,NCH,2]  1 MB

    dim3 ggrid(M_ / 128, NTOT / 64);                  // 256 x 24
    ssm_gemm_wmma<<<ggrid, 256, 0, stream>>>(u, Wproj, Wres, xr, yres);

    const int nChunkThreads = Bb * Rr * NCH;          // 131072
    ssm_scan_local<<<nChunkThreads / 256, 256, 0, stream>>>(
        xr, lraw, omega, h0r, h0i, yres, hend);
    ssm_scan_carry<<<(Bb * Rr) / 256, 256, 0, stream>>>(
        lraw, omega, hend, carry, finals);
    ssm_scan_apply<<<nChunkThreads / 256, 256, 0, stream>>>(
        lraw, omega, carry, yres);
    ssm_layernorm<<<M_ / 8, 256, 0, stream>>>(yres, gamma, beta);
}